// GCNPipeline_41034117546446
// MI455X (gfx1250) — compile-verified
//
#include <hip/hip_runtime.h>
#include <hip/hip_bf16.h>

typedef __attribute__((ext_vector_type(2))) float v2f;
typedef __attribute__((ext_vector_type(8))) float v8f;

#define N_NODES_C 50000
#define N_EDGES_C 800000
#define IN_DIM_C  128
#define HID_C     96
#define ATTN_HID_C 32
#define OUT_DIM_C 16
#define N_SPOTS_C 5000
#define EPS_C 1e-5f

// ---------------------------------------------------------------------------
// Generic zero kernel
// ---------------------------------------------------------------------------
__global__ void zero_kernel(float* p, int n) {
    int i = blockIdx.x * blockDim.x + threadIdx.x;
    if (i < n) p[i] = 0.0f;
}

// ---------------------------------------------------------------------------
// WMMA f32 GEMM: C[M,N] = A[M,K] @ W[K,N] (+bias, optional ReLU)
// One wave computes a 16x16 tile using V_WMMA_F32_16X16X4_F32.
// Requires: K % 8 == 0, N % 16 == 0. M tail rows handled by clamp+store-guard.
// ---------------------------------------------------------------------------
__global__ void gemm_wmma_kernel(const float* __restrict__ A,
                                 const float* __restrict__ W,
                                 const float* __restrict__ bias,
                                 float* __restrict__ Cout,
                                 int M, int K, int N, int relu)
{
    int wave = (blockIdx.x * blockDim.x + threadIdx.x) >> 5;
    int lane = threadIdx.x & 31;
    int nTiles = N >> 4;
    int mTiles = (M + 15) >> 4;
    if (wave >= mTiles * nTiles) return;

    int mT = wave / nTiles;
    int nT = wave - mT * nTiles;

    int row  = mT * 16 + (lane & 15);
    int rrow = row < M ? row : (M - 1);      // clamp: garbage rows never stored
    int col  = nT * 16 + (lane & 15);
    int kOff = (lane >> 4) << 1;             // lanes 16-31 hold K+2,K+3

    v8f acc0 = {};
    v8f acc1 = {};
    const float* arow = A + (size_t)rrow * K;

    for (int k = 0; k < K; k += 8) {
        int ka = k + kOff;
        v2f a0 = *(const v2f*)(arow + ka);
        v2f b0;
        b0.x = W[(size_t)ka * N + col];
        b0.y = W[(size_t)(ka + 1) * N + col];
        acc0 = __builtin_amdgcn_wmma_f32_16x16x4_f32(false, a0, false, b0,
                                                     (short)0, acc0, false, false);
        int kb = ka + 4;
        v2f a1 = *(const v2f*)(arow + kb);
        v2f b1;
        b1.x = W[(size_t)kb * N + col];
        b1.y = W[(size_t)(kb + 1) * N + col];
        acc1 = __builtin_amdgcn_wmma_f32_16x16x4_f32(false, a1, false, b1,
                                                     (short)0, acc1, false, false);
    }
    v8f acc = acc0 + acc1;

    float bv = bias ? bias[col] : 0.0f;
    int baseRow = mT * 16 + ((lane >> 4) << 3);   // lanes 16-31 hold rows M+8..M+15
#pragma unroll
    for (int r = 0; r < 8; ++r) {
        int orow = baseRow + r;
        if (orow < M) {
            float v = acc[r] + bv;
            if (relu) v = fmaxf(v, 0.0f);
            Cout[(size_t)orow * N + col] = v;
        }
    }
}

// ---------------------------------------------------------------------------
// GraphNorm: per-column sum and sum-of-squares (N fixed = 96)
// ---------------------------------------------------------------------------
__global__ void colstats_kernel(const float* __restrict__ X, int M,
                                float* __restrict__ sum, float* __restrict__ sumsq)
{
    int c = threadIdx.x;                     // blockDim.x == 96
    float s = 0.0f, q = 0.0f;
    for (int r = blockIdx.x; r < M; r += gridDim.x) {
        float v = X[(size_t)r * HID_C + c];
        s += v;
        q += v * v;
    }
    atomicAdd(&sum[c], s);
    atomicAdd(&sumsq[c], q);
}

// var = E[x^2] + (a^2 - 2a) * mean^2   (derived from centered = x - a*mean)
__global__ void gn_apply_kernel(float* __restrict__ X, int M,
                                const float* __restrict__ sum,
                                const float* __restrict__ sumsq,
                                const float* __restrict__ w,
                                const float* __restrict__ b,
                                const float* __restrict__ a,
                                int relu)
{
    int idx = blockIdx.x * blockDim.x + threadIdx.x;
    if (idx >= M * HID_C) return;
    int c = idx % HID_C;
    float invM = 1.0f / (float)M;
    float m  = sum[c] * invM;
    float av = a[c];
    float var = sumsq[c] * invM + (av * av - 2.0f * av) * m * m;
    float v = (X[idx] - av * m) * rsqrtf(var + EPS_C) * w[c] + b[c];
    if (relu) v = fmaxf(v, 0.0f);
    X[idx] = v;
}

// ---------------------------------------------------------------------------
// Degrees + symmetric norm
// ---------------------------------------------------------------------------
__global__ void deg_kernel(const int* __restrict__ dst, int E, float* __restrict__ deg) {
    int e = blockIdx.x * blockDim.x + threadIdx.x;
    if (e < E) atomicAdd(&deg[dst[e]], 1.0f);
}

__global__ void dinv_kernel(float* __restrict__ deg, int n) {
    int i = blockIdx.x * blockDim.x + threadIdx.x;
    if (i < n) deg[i] = rsqrtf(deg[i] + 1.0f);   // +1 for self loop
}

// agg = dinv^2 * hW + bias  (self-loop message + gcn bias, fully initializes agg)
__global__ void init_agg_kernel(const float* __restrict__ hw,
                                const float* __restrict__ dinv,
                                const float* __restrict__ bias,
                                float* __restrict__ agg, int M)
{
    int idx = blockIdx.x * blockDim.x + threadIdx.x;
    if (idx >= M * HID_C) return;
    int r = idx / HID_C;
    int c = idx - r * HID_C;
    float di = dinv[r];
    agg[idx] = di * di * hw[idx] + bias[c];
}

// One wave per edge; 3 features per lane (96 = 3*32). L2-resident gather+atomic.
__global__ void edge_scatter_kernel(const int* __restrict__ src,
                                    const int* __restrict__ dst,
                                    const float* __restrict__ dinv,
                                    const float* __restrict__ hw,
                                    float* __restrict__ agg, int E)
{
    int t = blockIdx.x * blockDim.x + threadIdx.x;
    int e = t >> 5;
    int lane = t & 31;
    if (e >= E) return;
    int s = src[e];
    int d = dst[e];
    float nrm = dinv[s] * dinv[d];
    const float* hrow = hw + (size_t)s * HID_C;
    float* arow = agg + (size_t)d * HID_C;
#pragma unroll
    for (int i = 0; i < 3; ++i) {
        int f = lane + 32 * i;
        atomicAdd(&arow[f], hrow[f] * nrm);
    }
}

// ---------------------------------------------------------------------------
// Attention scores: scores[n] = dot(attn1[n,0:32], W2) + b2 ; wave reduction
// ---------------------------------------------------------------------------
__global__ void scores_kernel(const float* __restrict__ attn1,
                              const float* __restrict__ W2,
                              const float* __restrict__ b2,
                              float* __restrict__ scores, int M)
{
    int wave = (blockIdx.x * blockDim.x + threadIdx.x) >> 5;
    int lane = threadIdx.x & 31;
    if (wave >= M) return;
    float v = attn1[(size_t)wave * ATTN_HID_C + lane] * W2[lane];
#pragma unroll
    for (int off = 16; off >= 1; off >>= 1) v += __shfl_xor(v, off, 32);
    if (lane == 0) scores[wave] = v + b2[0];
}

// monotonic uint key for float atomicMax
__device__ __forceinline__ unsigned fkey(float f) {
    unsigned u = __float_as_uint(f);
    return (u & 0x80000000u) ? ~u : (u | 0x80000000u);
}
__device__ __forceinline__ float unfkey(unsigned k) {
    unsigned u = (k & 0x80000000u) ? (k & 0x7FFFFFFFu) : ~k;
    return __uint_as_float(u);
}

__global__ void smax_kernel(const float* __restrict__ scores,
                            const int* __restrict__ seg,
                            unsigned* __restrict__ smax, int M)
{
    int i = blockIdx.x * blockDim.x + threadIdx.x;
    if (i < M) atomicMax(&smax[seg[i]], fkey(scores[i]));
}

__global__ void expdenom_kernel(const float* __restrict__ scores,
                                const int* __restrict__ seg,
                                const unsigned* __restrict__ smax,
                                float* __restrict__ e,
                                float* __restrict__ denom, int M)
{
    int i = blockIdx.x * blockDim.x + threadIdx.x;
    if (i >= M) return;
    int s = seg[i];
    float x = __expf(scores[i] - unfkey(smax[s]));
    e[i] = x;
    atomicAdd(&denom[s], x);
}

// spot[seg[n], :] += h[n, :] * (e[n] / denom[seg[n]]) ; one wave per node
__global__ void spot_scatter_kernel(const float* __restrict__ h,
                                    const int* __restrict__ seg,
                                    const float* __restrict__ e,
                                    const float* __restrict__ denom,
                                    float* __restrict__ spot, int M)
{
    int wave = (blockIdx.x * blockDim.x + threadIdx.x) >> 5;
    int lane = threadIdx.x & 31;
    if (wave >= M) return;
    int s = seg[wave];
    float w = e[wave] / denom[s];
    const float* hrow = h + (size_t)wave * HID_C;
    float* srow = spot + (size_t)s * HID_C;
#pragma unroll
    for (int i = 0; i < 3; ++i) {
        int f = lane + 32 * i;
        atomicAdd(&srow[f], hrow[f] * w);
    }
}

// ---------------------------------------------------------------------------
// Host-side helpers
// ---------------------------------------------------------------------------
static inline void launch_gemm(const float* A, const float* W, const float* bias,
                               float* C, int M, int K, int N, int relu, hipStream_t s)
{
    long long waves = (long long)((M + 15) / 16) * (N / 16);
    long long threads = waves * 32;
    int block = 256;
    int grid = (int)((threads + block - 1) / block);
    gemm_wmma_kernel<<<grid, block, 0, s>>>(A, W, bias, C, M, K, N, relu);
}

static inline void launch_zero(float* p, int n, hipStream_t s) {
    zero_kernel<<<(n + 255) / 256, 256, 0, s>>>(p, n);
}

static inline void graph_norm(float* X, int M, const float* w, const float* b,
                              const float* a, int relu, float* colsum, float* colsq,
                              hipStream_t s)
{
    launch_zero(colsum, 2 * HID_C, s);   // colsum and colsq are contiguous
    colstats_kernel<<<256, HID_C, 0, s>>>(X, M, colsum, colsq);
    gn_apply_kernel<<<(M * HID_C + 255) / 256, 256, 0, s>>>(X, M, colsum, colsq,
                                                            w, b, a, relu);
}

extern "C" void kernel_launch(void* const* d_in, const int* in_sizes, int n_in,
                              void* d_out, int out_size, void* d_ws, size_t ws_size,
                              hipStream_t stream)
{
    const float* x        = (const float*)d_in[0];
    const int*   eidx     = (const int*)  d_in[1];
    const int*   c2s      = (const int*)  d_in[2];
    // d_in[3] = num_spots (known constant N_SPOTS_C)
    const float* proj_W   = (const float*)d_in[4];
    const float* proj_b   = (const float*)d_in[5];
    const float* gn0_w    = (const float*)d_in[6];
    const float* gn0_b    = (const float*)d_in[7];
    const float* gn0_a    = (const float*)d_in[8];
    const float* gcn1_W   = (const float*)d_in[9];
    const float* gcn1_b   = (const float*)d_in[10];
    const float* gn1_w    = (const float*)d_in[11];
    const float* gn1_b    = (const float*)d_in[12];
    const float* gn1_a    = (const float*)d_in[13];
    const float* gcn2_W   = (const float*)d_in[14];
    const float* gcn2_b   = (const float*)d_in[15];
    const float* gn2_w    = (const float*)d_in[16];
    const float* gn2_b    = (const float*)d_in[17];
    const float* gn2_a    = (const float*)d_in[18];
    const float* attn_W1  = (const float*)d_in[19];
    const float* attn_b1  = (const float*)d_in[20];
    const float* attn_W2  = (const float*)d_in[21];
    const float* attn_b2  = (const float*)d_in[22];
    const float* mlp_W1   = (const float*)d_in[23];
    const float* mlp_b1   = (const float*)d_in[24];
    const float* mlp_gn_w = (const float*)d_in[25];
    const float* mlp_gn_b = (const float*)d_in[26];
    const float* mlp_gn_a = (const float*)d_in[27];
    const float* mlp_W2   = (const float*)d_in[28];
    const float* mlp_b2   = (const float*)d_in[29];

    const int* src = eidx;                 // edge_index[0, :]
    const int* dst = eidx + N_EDGES_C;     // edge_index[1, :]

    // ---- workspace layout (floats) ----
    float* ws = (float*)d_ws;
    const size_t NODEF = (size_t)N_NODES_C * HID_C;          // 4.8M
    float* A      = ws;                                      // [50000,96]
    float* B      = ws + NODEF;                              // [50000,96] (also attn1)
    float* C      = ws + 2 * NODEF;                          // [50000,96]
    float* small  = ws + 3 * NODEF;
    float* scores = small;                                   // 50000
    float* evals  = small + N_NODES_C;                       // 50000
    float* dinv   = small + 2 * N_NODES_C;                   // 50000 (deg then rsqrt)
    float* colsum = small + 3 * N_NODES_C;                   // 96 (+96 colsq contiguous)
    float* colsq  = colsum + HID_C;
    unsigned* smax = (unsigned*)(colsum + 2 * HID_C);        // 5000
    float* denom  = (float*)smax + N_SPOTS_C;                // 5000
    float* spot   = denom + N_SPOTS_C;                       // 5000*96
    float* z1     = spot + (size_t)N_SPOTS_C * HID_C;        // 5000*96

    const int NE = (int)(NODEF);                 // node-elem count
    (void)in_sizes; (void)n_in; (void)out_size; (void)ws_size;

    // 1) proj: x @ proj_W + b  -> A ; GraphNorm(gn0) + ReLU
    launch_gemm(x, proj_W, proj_b, A, N_NODES_C, IN_DIM_C, HID_C, 0, stream);
    graph_norm(A, N_NODES_C, gn0_w, gn0_b, gn0_a, 1, colsum, colsq, stream);

    // 2) degrees -> dinv
    launch_zero(dinv, N_NODES_C, stream);
    deg_kernel<<<(N_EDGES_C + 255) / 256, 256, 0, stream>>>(dst, N_EDGES_C, dinv);
    dinv_kernel<<<(N_NODES_C + 255) / 256, 256, 0, stream>>>(dinv, N_NODES_C);

    // 3) GCN layer 1: hW=B ; agg=C ; gn1+ReLU on C
    launch_gemm(A, gcn1_W, nullptr, B, N_NODES_C, HID_C, HID_C, 0, stream);
    init_agg_kernel<<<(NE + 255) / 256, 256, 0, stream>>>(B, dinv, gcn1_b, C, N_NODES_C);
    edge_scatter_kernel<<<(N_EDGES_C * 32 + 255) / 256, 256, 0, stream>>>(
        src, dst, dinv, B, C, N_EDGES_C);
    graph_norm(C, N_NODES_C, gn1_w, gn1_b, gn1_a, 1, colsum, colsq, stream);

    // 4) GCN layer 2: hW=B ; agg=A ; gn2+ReLU on A  (A = final h)
    launch_gemm(C, gcn2_W, nullptr, B, N_NODES_C, HID_C, HID_C, 0, stream);
    init_agg_kernel<<<(NE + 255) / 256, 256, 0, stream>>>(B, dinv, gcn2_b, A, N_NODES_C);
    edge_scatter_kernel<<<(N_EDGES_C * 32 + 255) / 256, 256, 0, stream>>>(
        src, dst, dinv, B, A, N_EDGES_C);
    graph_norm(A, N_NODES_C, gn2_w, gn2_b, gn2_a, 1, colsum, colsq, stream);

    // 5) attention scores: attn1 = relu(h @ W1 + b1) in B ; scores dot-reduce
    launch_gemm(A, attn_W1, attn_b1, B, N_NODES_C, HID_C, ATTN_HID_C, 1, stream);
    scores_kernel<<<(N_NODES_C * 32 + 255) / 256, 256, 0, stream>>>(
        B, attn_W2, attn_b2, scores, N_NODES_C);

    // 6) segment softmax + weighted segment sum into spot
    launch_zero((float*)smax, N_SPOTS_C, stream);
    launch_zero(denom, N_SPOTS_C, stream);
    launch_zero(spot, N_SPOTS_C * HID_C, stream);
    smax_kernel<<<(N_NODES_C + 255) / 256, 256, 0, stream>>>(scores, c2s, smax, N_NODES_C);
    expdenom_kernel<<<(N_NODES_C + 255) / 256, 256, 0, stream>>>(
        scores, c2s, smax, evals, denom, N_NODES_C);
    spot_scatter_kernel<<<(N_NODES_C * 32 + 255) / 256, 256, 0, stream>>>(
        A, c2s, evals, denom, spot, N_NODES_C);

    // 7) MLP head: z1 = spot @ W1 + b1 ; GraphNorm + ReLU ; out = z1 @ W2 + b2
    launch_gemm(spot, mlp_W1, mlp_b1, z1, N_SPOTS_C, HID_C, HID_C, 0, stream);
    graph_norm(z1, N_SPOTS_C, mlp_gn_w, mlp_gn_b, mlp_gn_a, 1, colsum, colsq, stream);
    launch_gemm(z1, mlp_W2, mlp_b2, (float*)d_out, N_SPOTS_C, HID_C, OUT_DIM_C, 0, stream);
}